// MeanAggregator_13108240187691
// MI455X (gfx1250) — compile-verified
//
#include <hip/hip_runtime.h>
#include <stdint.h>

#define NSAMP 12
#define DF 64
#define WAVES_PER_BLOCK 8
#define THREADS 256
#define NODES_PER_BLOCK 16   // 8 waves * 2 nodes per wave

typedef int   v4i __attribute__((ext_vector_type(4)));
typedef float v4f __attribute__((ext_vector_type(4)));

// Mean aggregation via CDNA5 async gather-to-LDS DMA, double-buffered per wave.
// Wave layout: lane = sel*16 + chunk ; sel in {0,1} picks node, chunk picks float4.
__global__ __launch_bounds__(THREADS) void
mean_agg_async(const float* __restrict__ feat,
               const int*   __restrict__ nbr,
               float*       __restrict__ out,
               int n_nodes, int iters, int node_stride)
{
    // [wave][buf][sel][slot][chunk] of v4f -> 8*2*2*12*16*16B = 96 KB
    __shared__ v4f tile[WAVES_PER_BLOCK][2][2][NSAMP][16];

    const int lane  = threadIdx.x & 31;
    const int wave  = threadIdx.x >> 5;
    const int sel   = lane >> 4;   // node within wave
    const int chunk = lane & 15;   // float4 column

    // LDS byte offset of this wave's staging region: generic shared pointer's
    // low 32 bits are the LDS offset (aperture layout, ISA 10.2).
    const uint32_t lds_wave = (uint32_t)(uintptr_t)(&tile[wave][0][0][0][0]);
    const uint32_t lds_lane = lds_wave
                            + (uint32_t)sel   * (uint32_t)(NSAMP * 16 * 16)
                            + (uint32_t)chunk * 16u;

    const int node0 = blockIdx.x * NODES_PER_BLOCK + wave * 2 + sel;
    const unsigned long long feat_base =
        (unsigned long long)(uintptr_t)feat + (unsigned long long)chunk * 16ull;

    // Issue 12 async gather loads (one row-chunk per slot) into buffer `buf`.
    auto issue = [&](int it, int buf) {
        int node = node0 + it * node_stride;
        if (node > n_nodes - 1) node = n_nodes - 1;   // clamp: EXEC stays full
        const v4i* ip = (const v4i*)(nbr + (size_t)node * NSAMP); // 48B row, 16B aligned
        v4i i0 = __builtin_nontemporal_load(ip + 0);
        v4i i1 = __builtin_nontemporal_load(ip + 1);
        v4i i2 = __builtin_nontemporal_load(ip + 2);
        int idx[NSAMP] = { i0.x, i0.y, i0.z, i0.w,
                           i1.x, i1.y, i1.z, i1.w,
                           i2.x, i2.y, i2.z, i2.w };
        const uint32_t lb = lds_lane + (uint32_t)buf * (uint32_t)(2 * NSAMP * 16 * 16);
#pragma unroll
        for (int s = 0; s < NSAMP; ++s) {
            unsigned long long ga =
                feat_base + (unsigned long long)(unsigned)idx[s] * 256ull;
            uint32_t l = lb + (uint32_t)s * 256u;
            asm volatile("global_load_async_to_lds_b128 %0, %1, off"
                         :: "v"(l), "v"(ga) : "memory");
        }
    };

    issue(0, 0);
    for (int it = 0; it < iters; ++it) {
        const int buf = it & 1;
        if (it + 1 < iters) {
            issue(it + 1, buf ^ 1);                          // prefetch next tile
            asm volatile("s_wait_asynccnt 12" ::: "memory"); // current tile done
        } else {
            asm volatile("s_wait_asynccnt 0" ::: "memory");
        }

        v4f acc = tile[wave][buf][sel][0][chunk];
#pragma unroll
        for (int s = 1; s < NSAMP; ++s) {
            acc += tile[wave][buf][sel][s][chunk];
        }
        acc *= (1.0f / (float)NSAMP);

        const int node = node0 + it * node_stride;
        if (node < n_nodes) {
            v4f* op = (v4f*)(out + (size_t)node * DF) + chunk;
            __builtin_nontemporal_store(acc, op);   // write-once stream, keep L2 for feat
        }
    }
}

extern "C" void kernel_launch(void* const* d_in, const int* in_sizes, int n_in,
                              void* d_out, int out_size, void* d_ws, size_t ws_size,
                              hipStream_t stream) {
    const float* feat = (const float*)d_in[0];
    const int*   nbr  = (const int*)d_in[1];
    float*       out  = (float*)d_out;

    const int n_nodes = in_sizes[1] / NSAMP;   // 100000 for the reference shapes

    const int iters = 10;   // pipeline depth per wave (double-buffered)
    int blocks = (n_nodes + NODES_PER_BLOCK * iters - 1) / (NODES_PER_BLOCK * iters);
    if (blocks < 1) blocks = 1;                // 625 blocks for n_nodes=100000 (exact)
    const int node_stride = blocks * NODES_PER_BLOCK;

    mean_agg_async<<<blocks, THREADS, 0, stream>>>(feat, nbr, out,
                                                   n_nodes, iters, node_stride);
}